// GCNNet_19464791786080
// MI455X (gfx1250) — compile-verified
//
#include <hip/hip_runtime.h>

// ---------------------------------------------------------------------------
// GCN forward for MI455X (gfx1250, wave32).
//   layer: h' = relu( Dinv (A+I) Dinv (h W) + b ),  3 layers, concat, linear,
//   log_softmax.  GEMMs use V_WMMA_F32_16X16X4_F32 (fp32 matrix core path).
//   B is staged in LDS K-pair-interleaved so each fragment is one ds_load_b64
//   into an even-aligned VGPR pair (no shuffle movs before the WMMA).
// ---------------------------------------------------------------------------

typedef __attribute__((ext_vector_type(2))) float v2f;
typedef __attribute__((ext_vector_type(8))) float v8f;

static constexpr int GN   = 100000;  // nodes
static constexpr int GE   = 1000000; // edges
static constexpr int DIN  = 128;
static constexpr int HID  = 64;
static constexpr int CAT  = 192;     // 3*HID
static constexpr int LAB  = 40;

// padded LDS row stride (in float2 units): WP % 32 == 16  =>  lane halves
// (one kp-row apart = WP*8 bytes) hit disjoint 32-bank groups for b64 loads.
template <int W> struct PadRow {
    static constexpr int value = W + ((48 - (W % 32)) % 32);
};

// ---------------------------------------------------------------------------
// WMMA fp32 GEMM:  out[M x dout] = A[M x K] * B[K x dout]
//   NT = number of 16-wide N tiles (dout_pad = NT*16 >= dout), K even.
//   8 waves / block, each wave owns a 16-row stripe across all NT tiles.
//   Epilogue (fused):
//     agg != nullptr :  out[m][n] = val ; agg[m][n] = bias[n] + dinv[m]^2*val
//     agg == nullptr :  out[m][n] = val + bias[n]
// ---------------------------------------------------------------------------
template <int NT>
__launch_bounds__(256)
__global__ void gemm_wmma_f32(const float* __restrict__ A, int lda,
                              const float* __restrict__ B, int K, int dout,
                              int M, float* __restrict__ out, int ldo,
                              const float* __restrict__ bias, int bias_to_out,
                              float* __restrict__ agg,
                              const float* __restrict__ dinv)
{
    constexpr int W  = NT * 16;
    constexpr int WP = PadRow<W>::value;   // float2 stride per kp-row
    extern __shared__ v2f Bs2[];           // [K/2][WP] pairs (B[2kp][n],B[2kp+1][n])

    // stage B pairs (zero-pad columns >= dout)
    const int KP = K >> 1;
    for (int idx = threadIdx.x; idx < KP * W; idx += 256) {
        int kp = idx / W;
        int n  = idx - kp * W;
        v2f p;
        p.x = (n < dout) ? B[(2 * kp)     * dout + n] : 0.0f;
        p.y = (n < dout) ? B[(2 * kp + 1) * dout + n] : 0.0f;
        Bs2[kp * WP + n] = p;
    }
    __syncthreads();

    const int wave = threadIdx.x >> 5;
    const int lane = threadIdx.x & 31;
    const int l15  = lane & 15;
    const int half = lane >> 4;        // 0: K base 0 / M base 0, 1: K base 2 / M base 8
    const int kb   = half * 2;

    const int m0 = (blockIdx.x * 8 + wave) * 16;
    int mA = m0 + l15;                 // A row this lane feeds
    if (mA > M - 1) mA = M - 1;        // clamp (branchless tail handling)
    const float* Arow = A + (size_t)mA * (size_t)lda;

    v8f acc[NT];
#pragma unroll
    for (int t = 0; t < NT; ++t)
        acc[t] = (v8f){0.f, 0.f, 0.f, 0.f, 0.f, 0.f, 0.f, 0.f};

    for (int k = 0; k < K; k += 4) {
        // A fragment 16x4: lane holds A[m][k+kb], A[m][k+kb+1]  (8B aligned)
        v2f a = *(const v2f*)(Arow + k + kb);
        // kp-row for this lane half: (k+kb)/2 = k/2 + half
        const v2f* brow = Bs2 + ((k >> 1) + half) * WP + l15;
#pragma unroll
        for (int t = 0; t < NT; ++t) {
            v2f b = brow[t * 16];      // single ds_load_b64, aligned pair
            acc[t] = __builtin_amdgcn_wmma_f32_16x16x4_f32(
                false, a, false, b, (short)0, acc[t], false, false);
        }
    }

    // epilogue
#pragma unroll
    for (int t = 0; t < NT; ++t) {
        const int n = t * 16 + l15;
        float bv = 0.0f;
        if (bias && n < dout) bv = bias[n];
#pragma unroll
        for (int v = 0; v < 8; ++v) {
            const int m = m0 + half * 8 + v;
            if (m < M && n < dout) {
                const float val = acc[t][v];
                if (agg) {
                    out[(size_t)m * ldo + n] = val;
                    const float di = dinv[m];
                    agg[(size_t)m * dout + n] = bv + di * di * val;
                } else {
                    out[(size_t)m * ldo + n] = val + (bias_to_out ? bv : 0.0f);
                }
            }
        }
    }
}

// ---------------------------------------------------------------------------
// degree / dinv
// ---------------------------------------------------------------------------
__global__ void deg_init(float* __restrict__ d, int M) {
    int i = blockIdx.x * blockDim.x + threadIdx.x;
    if (i < M) d[i] = 1.0f;            // self loop
}

__global__ void deg_scatter(const int* __restrict__ ei, int E,
                            float* __restrict__ deg) {
    int e = blockIdx.x * blockDim.x + threadIdx.x;
    if (e < E) unsafeAtomicAdd(&deg[ei[E + e]], 1.0f);   // dst row
}

__global__ void deg_rsqrt(float* __restrict__ d, int M) {
    int i = blockIdx.x * blockDim.x + threadIdx.x;
    if (i < M) d[i] = rsqrtf(d[i]);
}

// ---------------------------------------------------------------------------
// edge scatter: one wave32 per edge, lane c handles channels 2c, 2c+1
//   agg[dst] += dinv[src]*dinv[dst] * hw[src]
// ---------------------------------------------------------------------------
__global__ void edge_scatter(const int* __restrict__ ei, int E,
                             const float* __restrict__ dinv,
                             const float* __restrict__ hw,
                             float* __restrict__ agg)
{
    int gid = blockIdx.x * blockDim.x + threadIdx.x;
    int e = gid >> 5;
    if (e >= E) return;
    int lane = gid & 31;
    int s = ei[e];
    int d = ei[E + e];
    float w = dinv[s] * dinv[d];
    int c = lane * 2;
    float2 v = *(const float2*)(hw + (size_t)s * HID + c);
    float* p = agg + (size_t)d * HID + c;
    unsafeAtomicAdd(p,     w * v.x);
    unsafeAtomicAdd(p + 1, w * v.y);
}

// ---------------------------------------------------------------------------
// relu + write into the [N,192] concat buffer (float4 vectorized)
// ---------------------------------------------------------------------------
__global__ void relu_concat(const float* __restrict__ agg,
                            float* __restrict__ Hcat, int M, int col0)
{
    int gid = blockIdx.x * blockDim.x + threadIdx.x;
    int i = gid >> 4;                  // 16 threads per row (64 ch / 4)
    if (i >= M) return;
    int c = (gid & 15) * 4;
    float4 v = *(const float4*)(agg + (size_t)i * HID + c);
    v.x = fmaxf(v.x, 0.0f);
    v.y = fmaxf(v.y, 0.0f);
    v.z = fmaxf(v.z, 0.0f);
    v.w = fmaxf(v.w, 0.0f);
    *(float4*)(Hcat + (size_t)i * CAT + col0 + c) = v;
}

// ---------------------------------------------------------------------------
// in-place log_softmax over 40 labels, one thread per row
// ---------------------------------------------------------------------------
__global__ void logsoftmax40(float* __restrict__ out, int M)
{
    int i = blockIdx.x * blockDim.x + threadIdx.x;
    if (i >= M) return;
    float* row = out + (size_t)i * LAB;
    float v[LAB];
    float mx = -3.4e38f;
#pragma unroll
    for (int c = 0; c < LAB; ++c) { v[c] = row[c]; mx = fmaxf(mx, v[c]); }
    float s = 0.0f;
#pragma unroll
    for (int c = 0; c < LAB; ++c) s += __expf(v[c] - mx);
    float ls = __logf(s);
#pragma unroll
    for (int c = 0; c < LAB; ++c) row[c] = v[c] - mx - ls;
}

// ---------------------------------------------------------------------------
// launcher
// ---------------------------------------------------------------------------
extern "C" void kernel_launch(void* const* d_in, const int* in_sizes, int n_in,
                              void* d_out, int out_size, void* d_ws, size_t ws_size,
                              hipStream_t stream)
{
    (void)in_sizes; (void)n_in; (void)out_size; (void)ws_size;

    const float* x  = (const float*)d_in[0];
    const int*   ei = (const int*)d_in[1];
    const float* Wm[3] = {(const float*)d_in[2], (const float*)d_in[4],
                          (const float*)d_in[6]};
    const float* bm[3] = {(const float*)d_in[3], (const float*)d_in[5],
                          (const float*)d_in[7]};
    const float* Wl = (const float*)d_in[8];
    const float* bl = (const float*)d_in[9];
    float* out = (float*)d_out;

    // workspace carve-up (256B aligned)
    char* ws = (char*)d_ws;
    auto align256 = [](size_t b) { return (b + 255) & ~(size_t)255; };
    size_t off = 0;
    float* dinv = (float*)(ws + off); off += align256((size_t)GN * 4);
    float* hw   = (float*)(ws + off); off += align256((size_t)GN * HID * 4);
    float* agg  = (float*)(ws + off); off += align256((size_t)GN * HID * 4);
    float* Hcat = (float*)(ws + off); off += align256((size_t)GN * CAT * 4);

    const dim3 blk(256);

    // degree -> dinv
    deg_init   <<<(GN + 255) / 256, blk, 0, stream>>>(dinv, GN);
    deg_scatter<<<(GE + 255) / 256, blk, 0, stream>>>(ei, GE, dinv);
    deg_rsqrt  <<<(GN + 255) / 256, blk, 0, stream>>>(dinv, GN);

    const int gemmBlocks = (GN + 127) / 128;   // 8 waves x 16 rows per block

    for (int l = 0; l < 3; ++l) {
        const float* Ain = (l == 0) ? x : (Hcat + HID * (l - 1));
        const int lda = (l == 0) ? DIN : CAT;
        const int K   = (l == 0) ? DIN : HID;
        const size_t shmem = (size_t)(K / 2) * PadRow<64>::value * sizeof(float2);
        // hw = h @ W ; fused: agg = b + dinv^2 * hw  (self-loop term)
        gemm_wmma_f32<4><<<gemmBlocks, blk, shmem, stream>>>(
            Ain, lda, Wm[l], K, HID, GN, hw, HID, bm[l], 0, agg, dinv);
        // agg[dst] += dinv_s*dinv_d * hw[src]  over all edges
        edge_scatter<<<(GE * 32 + 255) / 256, blk, 0, stream>>>(
            ei, GE, dinv, hw, agg);
        // Hcat[:, 64l:64l+64] = relu(agg)
        relu_concat<<<((size_t)GN * 16 + 255) / 256, blk, 0, stream>>>(
            agg, Hcat, GN, HID * l);
    }

    // logits = Hcat @ Wl + bl   (40 cols, padded to 48 = 3 N-tiles)
    const size_t shmemF = (size_t)(CAT / 2) * PadRow<48>::value * sizeof(float2);
    gemm_wmma_f32<3><<<gemmBlocks, blk, shmemF, stream>>>(
        Hcat, CAT, Wl, CAT, LAB, GN, out, LAB, bl, 1, nullptr, nullptr);

    // in-place log_softmax
    logsoftmax40<<<(GN + 255) / 256, blk, 0, stream>>>(out, GN);
}